// OFTParametrization_29463475650678
// MI455X (gfx1250) — compile-verified
//
#include <hip/hip_runtime.h>
#include <math.h>
#include <stdint.h>

// ---------------------------------------------------------------------------
// OFT / COFT parametrization on MI455X (gfx1250, wave32, WMMA)
//   S = 0.5*scale*(R - R^T)          (scale from COFT norm projection)
//   inv(I+S) via Newton-Schulz:  X <- X (2I - (I+S) X),  X0 = I - S
//   Q = (I - S) X
//   filt_t[r*B+i][o] = sum_j Q_r[i][j] * W[o][r*B+j]
//   out = x @ filt_t        <- dominant GEMM: async-LDS double-buffered
// All GEMMs through V_WMMA_F32_16X16X4_F32 (fp32 matrix pipe).
// ---------------------------------------------------------------------------

#define IN_F   4096
#define OUT_F  4096
#define NBLK   4
#define BLKD   1024
#define MX     8192          // 8*1024 rows of x

#define WG_M   128
#define WG_N   64
#define KT     32

#define A_STRIDE 36          // padded LDS row stride for A (floats)
#define B_STRIDE 72          // padded LDS row stride for B (floats)

typedef __attribute__((ext_vector_type(2))) float v2f;
typedef __attribute__((ext_vector_type(8))) float v8f;

#if defined(__gfx1250__) && __has_builtin(__builtin_amdgcn_wmma_f32_16x16x4_f32)
#define HAVE_WMMA_F32 1
#else
#define HAVE_WMMA_F32 0
#endif

#if defined(__gfx1250__)
#define HAVE_ASYNC_LDS 1
#else
#define HAVE_ASYNC_LDS 0
#endif

#if HAVE_ASYNC_LDS
// GLOBAL_LOAD_ASYNC_TO_LDS_B128 via inline asm (clang builtin's pointer params
// use HIP language address spaces that aren't spellable from HIP source).
// VDST = 32-bit LDS byte address (low 32 bits of a flat LDS pointer, ISA 10.2),
// ADDR = 64-bit global address VGPR pair, no SADDR ("off").
__device__ __forceinline__ void async_load16(const void* g, void* l) {
    const unsigned int       lofs = (unsigned int)(uintptr_t)l;
    const unsigned long long ga   = (unsigned long long)(uintptr_t)g;
    asm volatile("global_load_async_to_lds_b128 %0, %1, off"
                 :: "v"(lofs), "v"(ga) : "memory");
}
__device__ __forceinline__ void wait_async0() {
#if __has_builtin(__builtin_amdgcn_s_wait_asynccnt)
    __builtin_amdgcn_s_wait_asynccnt(0);
#else
    asm volatile("s_wait_asynccnt 0" ::: "memory");
#endif
}
#endif

// ---------------------------------------------------------------------------
// Frobenius norm^2 per diagonal block
// ---------------------------------------------------------------------------
__global__ __launch_bounds__(256)
void oft_block_norm2(const float* __restrict__ Rm, float* __restrict__ norm2)
{
    __shared__ float red[256];
    const int r = blockIdx.y;
    const float* base = Rm + (long)r * BLKD * BLKD;
    float s = 0.f;
    for (long i = (long)blockIdx.x * blockDim.x + threadIdx.x;
         i < (long)BLKD * BLKD;
         i += (long)gridDim.x * blockDim.x) {
        float v = base[i];
        s += v * v;
    }
    red[threadIdx.x] = s;
    __syncthreads();
    for (int off = 128; off > 0; off >>= 1) {
        if ((int)threadIdx.x < off) red[threadIdx.x] += red[threadIdx.x + off];
        __syncthreads();
    }
    if (threadIdx.x == 0) atomicAdd(&norm2[r], red[0]);
}

// ---------------------------------------------------------------------------
// S = 0.5*scale*(R - R^T);  X0 = I - S   (Newton seed = 2I - (I+S))
// ---------------------------------------------------------------------------
__global__ __launch_bounds__(256)
void oft_build_skew(const float* __restrict__ Rm, const float* __restrict__ norm2,
                    float* __restrict__ S, float* __restrict__ X0)
{
    const int r = blockIdx.y;
    const long idx = (long)blockIdx.x * blockDim.x + threadIdx.x;  // 0..BLKD*BLKD-1
    const int i = (int)(idx >> 10);
    const int j = (int)(idx & (BLKD - 1));
    // eps = (1e-5 * 1024*1024) / sqrt(4)
    const float eps = 5.24288f;
    const float n = sqrtf(norm2[r]);
    const float scale = (n <= eps) ? 1.0f : (eps / n);
    const float* base = Rm + (long)r * BLKD * BLKD;
    const float a = base[(long)i * BLKD + j];
    const float b = base[(long)j * BLKD + i];
    const float sv = 0.5f * scale * (a - b);
    const long o = (long)r * BLKD * BLKD + idx;
    S[o]  = sv;
    X0[o] = ((i == j) ? 1.0f : 0.0f) - sv;
}

// ---------------------------------------------------------------------------
// Generic fp32 WMMA GEMM with epilogue modes (Newton / Cayley / filt stages):
//   aMode: 0 = A, 1 = I + A, 2 = I - A
//   bMode: 0 = B, 1 = 2I - B
//   bTrans: 0 = B row-major (K x N, ldb), 1 = element (k,n) = B[n*ldb + k]
// ---------------------------------------------------------------------------
__global__ __launch_bounds__(256)
void wmma_gemm_f32(const float* __restrict__ Abase, const float* __restrict__ Bbase,
                   float* __restrict__ Cbase,
                   int M, int N, int K, int lda, int ldb, int ldc,
                   long aBatch, long bBatch, long cBatch,
                   int aMode, int bMode, int bTrans)
{
    __shared__ float As[KT][WG_M + 1];
    __shared__ float Bs[KT][WG_N + 1];

    const int tid  = threadIdx.x;
    const int lane = tid & 31;
    const int wave = tid >> 5;          // 0..7
    const int wm   = (wave & 3) * 32;   // wave M offset inside WG tile
    const int wn   = (wave >> 2) * 32;  // wave N offset inside WG tile

    const int m0 = blockIdx.y * WG_M;
    const int n0 = blockIdx.x * WG_N;
    const int b  = blockIdx.z;

    const float* A = Abase + (long)b * aBatch;
    const float* B = Bbase + (long)b * bBatch;
    float*       C = Cbase + (long)b * cBatch;

    const int halfSel = lane >> 4;      // K half select (lanes 16-31 -> K+2)
    const int l16     = lane & 15;

    v8f acc[2][2];
    {
        v8f z = {};
        acc[0][0] = z; acc[0][1] = z; acc[1][0] = z; acc[1][1] = z;
    }

    for (int k0 = 0; k0 < K; k0 += KT) {
        // ---- stage A tile (WG_M x KT), store transposed As[k][m] ----
        #pragma unroll
        for (int p = 0; p < 4; ++p) {
            const int idx = tid + p * 256;       // 0..1023
            const int row = idx >> 3;            // 0..127
            const int kv  = (idx & 7) * 4;       // 0..28
            const int gm  = m0 + row;
            const int gk  = k0 + kv;
            const float4 v = *reinterpret_cast<const float4*>(A + (long)gm * lda + gk);
            float vv[4] = {v.x, v.y, v.z, v.w};
            #pragma unroll
            for (int q = 0; q < 4; ++q) {
                float a = vv[q];
                const int gkk = gk + q;
                if (aMode == 1)      a = a + ((gm == gkk) ? 1.0f : 0.0f); // I + A
                else if (aMode == 2) a = ((gm == gkk) ? 1.0f : 0.0f) - a; // I - A
                As[kv + q][row] = a;
            }
        }
        // ---- stage B tile (KT x WG_N) ----
        if (!bTrans) {
            #pragma unroll
            for (int p = 0; p < 2; ++p) {
                const int idx = tid + p * 256;   // 0..511
                const int row = idx >> 4;        // k: 0..31
                const int nv  = (idx & 15) * 4;  // n: 0..60
                const int gk  = k0 + row;
                const int gn  = n0 + nv;
                const float4 v = *reinterpret_cast<const float4*>(B + (long)gk * ldb + gn);
                float vv[4] = {v.x, v.y, v.z, v.w};
                #pragma unroll
                for (int q = 0; q < 4; ++q) {
                    float bv = vv[q];
                    const int gnn = gn + q;
                    if (bMode == 1) bv = ((gk == gnn) ? 2.0f : 0.0f) - bv;  // 2I - B
                    Bs[row][nv + q] = bv;
                }
            }
        } else {
            #pragma unroll
            for (int p = 0; p < 2; ++p) {
                const int idx = tid + p * 256;   // 0..511
                const int nr  = idx >> 3;        // n: 0..63
                const int kv  = (idx & 7) * 4;   // k: 0..28
                const int gn  = n0 + nr;
                const int gk  = k0 + kv;
                const float4 v = *reinterpret_cast<const float4*>(B + (long)gn * ldb + gk);
                Bs[kv + 0][nr] = v.x;
                Bs[kv + 1][nr] = v.y;
                Bs[kv + 2][nr] = v.z;
                Bs[kv + 3][nr] = v.w;
            }
        }
        if (k0 + KT < K) {
            __builtin_prefetch(A + (long)(m0 + (tid >> 1)) * lda + (k0 + KT), 0, 1);
        }
        __syncthreads();

        // ---- compute: 8 K-steps of 4 through the fp32 matrix pipe ----
        #pragma unroll
        for (int ks = 0; ks < KT; ks += 4) {
#if HAVE_WMMA_F32
            const int kk = ks + halfSel * 2;
            v2f afrag[2], bfrag[2];
            #pragma unroll
            for (int i = 0; i < 2; ++i) {
                const int m = wm + i * 16 + l16;
                afrag[i].x = As[kk][m];
                afrag[i].y = As[kk + 1][m];
            }
            #pragma unroll
            for (int j = 0; j < 2; ++j) {
                const int n = wn + j * 16 + l16;
                bfrag[j].x = Bs[kk][n];
                bfrag[j].y = Bs[kk + 1][n];
            }
            #pragma unroll
            for (int i = 0; i < 2; ++i)
                #pragma unroll
                for (int j = 0; j < 2; ++j)
                    acc[i][j] = __builtin_amdgcn_wmma_f32_16x16x4_f32(
                        false, afrag[i], false, bfrag[j],
                        (short)0, acc[i][j], false, false);
#else
            #pragma unroll
            for (int i = 0; i < 2; ++i)
                #pragma unroll
                for (int j = 0; j < 2; ++j) {
                    const int col = wn + j * 16 + l16;
                    #pragma unroll
                    for (int v = 0; v < 8; ++v) {
                        const int row = wm + i * 16 + v + halfSel * 8;
                        float s = acc[i][j][v];
                        #pragma unroll
                        for (int q = 0; q < 4; ++q)
                            s = fmaf(As[ks + q][row], Bs[ks + q][col], s);
                        acc[i][j][v] = s;
                    }
                }
#endif
        }
        __syncthreads();
    }

    // ---- store C per ISA 16x16 f32 C/D layout ----
    #pragma unroll
    for (int i = 0; i < 2; ++i) {
        #pragma unroll
        for (int j = 0; j < 2; ++j) {
            const int col = n0 + wn + j * 16 + l16;
            #pragma unroll
            for (int v = 0; v < 8; ++v) {
                const int row = m0 + wm + i * 16 + v + halfSel * 8;
                C[(long)row * ldc + col] = acc[i][j][v];
            }
        }
    }
}

// ---------------------------------------------------------------------------
// Plain fp32 WMMA GEMM, async global->LDS double buffered (final 0.27 TFLOP
// GEMM). LDS tiles kept in memory layout (DMA can't transform), padded:
//   A row stride 36 floats -> A-fragment banks (36m+k)%64 all distinct,
//     K-halves in disjoint mod-4 classes -> conflict-free.
//   B row stride 72 floats -> half-wave row offset 576B = +16 banks -> disjoint.
// Protocol per K-tile: issue async loads into buf^1, compute buf,
// s_wait_asynccnt 0, barrier, swap.
// ---------------------------------------------------------------------------
__global__ __launch_bounds__(256)
void wmma_gemm_f32_async(const float* __restrict__ A, const float* __restrict__ B,
                         float* __restrict__ C,
                         int M, int N, int K, int lda, int ldb, int ldc)
{
    __shared__ float As[2][WG_M][A_STRIDE];   // 2 * 128*36*4 = 36.0 KB
    __shared__ float Bs[2][KT][B_STRIDE];     // 2 *  32*72*4 = 18.0 KB

    const int tid  = threadIdx.x;
    const int lane = tid & 31;
    const int wave = tid >> 5;
    const int wm   = (wave & 3) * 32;
    const int wn   = (wave >> 2) * 32;

    const int m0 = blockIdx.y * WG_M;
    const int n0 = blockIdx.x * WG_N;

    const int halfSel = lane >> 4;
    const int l16     = lane & 15;

    v8f acc[2][2];
    {
        v8f z = {};
        acc[0][0] = z; acc[0][1] = z; acc[1][0] = z; acc[1][1] = z;
    }

    // Stage one K-tile (A: 1024 16B chunks, B: 512 16B chunks; 6 per thread).
    auto stage = [&](int buf, int k0) {
        #pragma unroll
        for (int p = 0; p < 4; ++p) {
            const int idx = tid + p * 256;     // 0..1023
            const int row = idx >> 3;          // m: 0..127
            const int kv  = (idx & 7) * 4;     // k: 0..28
            const float* g = A + (long)(m0 + row) * lda + (k0 + kv);
            float* l = &As[buf][row][kv];
#if HAVE_ASYNC_LDS
            async_load16(g, l);
#else
            *reinterpret_cast<float4*>(l) = *reinterpret_cast<const float4*>(g);
#endif
        }
        #pragma unroll
        for (int p = 0; p < 2; ++p) {
            const int idx = tid + p * 256;     // 0..511
            const int row = idx >> 4;          // k: 0..31
            const int nv  = (idx & 15) * 4;    // n: 0..60
            const float* g = B + (long)(k0 + row) * ldb + (n0 + nv);
            float* l = &Bs[buf][row][nv];
#if HAVE_ASYNC_LDS
            async_load16(g, l);
#else
            *reinterpret_cast<float4*>(l) = *reinterpret_cast<const float4*>(g);
#endif
        }
    };

    stage(0, 0);
#if HAVE_ASYNC_LDS
    wait_async0();
#endif
    __syncthreads();

    int buf = 0;
    for (int k0 = 0; k0 < K; k0 += KT) {
        // Prefetch next K-tile into the other buffer while computing this one.
        // Safe: the barrier ending iteration k0-KT guaranteed every wave is
        // done reading buf^1.
        if (k0 + KT < K) stage(buf ^ 1, k0 + KT);

        #pragma unroll
        for (int ks = 0; ks < KT; ks += 4) {
#if HAVE_WMMA_F32
            const int kk = ks + halfSel * 2;
            v2f afrag[2], bfrag[2];
            #pragma unroll
            for (int i = 0; i < 2; ++i) {
                const int m = wm + i * 16 + l16;
                afrag[i].x = As[buf][m][kk];
                afrag[i].y = As[buf][m][kk + 1];
            }
            #pragma unroll
            for (int j = 0; j < 2; ++j) {
                const int n = wn + j * 16 + l16;
                bfrag[j].x = Bs[buf][kk][n];
                bfrag[j].y = Bs[buf][kk + 1][n];
            }
            #pragma unroll
            for (int i = 0; i < 2; ++i)
                #pragma unroll
                for (int j = 0; j < 2; ++j)
                    acc[i][j] = __builtin_amdgcn_wmma_f32_16x16x4_f32(
                        false, afrag[i], false, bfrag[j],
                        (short)0, acc[i][j], false, false);
#else
            #pragma unroll
            for (int i = 0; i < 2; ++i)
                #pragma unroll
                for (int j = 0; j < 2; ++j) {
                    const int col = wn + j * 16 + l16;
                    #pragma unroll
                    for (int v = 0; v < 8; ++v) {
                        const int row = wm + i * 16 + v + halfSel * 8;
                        float s = acc[i][j][v];
                        #pragma unroll
                        for (int q = 0; q < 4; ++q)
                            s = fmaf(As[buf][row][ks + q], Bs[buf][ks + q][col], s);
                        acc[i][j][v] = s;
                    }
                }
#endif
        }

#if HAVE_ASYNC_LDS
        wait_async0();                         // next tile landed in LDS
#endif
        __syncthreads();                       // visible to all waves
        buf ^= 1;
    }

    #pragma unroll
    for (int i = 0; i < 2; ++i) {
        #pragma unroll
        for (int j = 0; j < 2; ++j) {
            const int col = n0 + wn + j * 16 + l16;
            #pragma unroll
            for (int v = 0; v < 8; ++v) {
                const int row = m0 + wm + i * 16 + v + halfSel * 8;
                C[(long)row * ldc + col] = acc[i][j][v];
            }
        }
    }
}

// ---------------------------------------------------------------------------
// Host-side pipeline
// ---------------------------------------------------------------------------
extern "C" void kernel_launch(void* const* d_in, const int* in_sizes, int n_in,
                              void* d_out, int out_size, void* d_ws, size_t ws_size,
                              hipStream_t stream)
{
    (void)in_sizes; (void)n_in; (void)out_size; (void)ws_size;

    const float* x    = (const float*)d_in[0];   // (8192, 4096)
    const float* oftR = (const float*)d_in[1];   // (4, 1024, 1024)
    const float* W    = (const float*)d_in[2];   // (4096, 4096)
    float*       out  = (float*)d_out;           // (8192, 4096)

    const long BB = (long)BLKD * BLKD;           // 1M elements per block
    float* ws    = (float*)d_ws;
    float* norm2 = ws;                           // 4 floats (padded to 64)
    float* S     = ws + 64;                      // 4 * BB
    float* Xa    = S  + NBLK * BB;               // 4 * BB
    float* Xb    = Xa + NBLK * BB;               // 4 * BB
    float* T     = Xb + NBLK * BB;               // 4 * BB (also holds Q)
    float* filt  = T  + NBLK * BB;               // 4096*4096 (filt_t, IN x OUT)

    (void)hipMemsetAsync(norm2, 0, NBLK * sizeof(float), stream);

    const dim3 thr(256);
    oft_block_norm2<<<dim3(256, NBLK), thr, 0, stream>>>(oftR, norm2);
    oft_build_skew<<<dim3(BLKD * BLKD / 256, NBLK), thr, 0, stream>>>(oftR, norm2, S, Xa);

    // Newton-Schulz: X <- X (2I - (I+S) X), 5 iterations (quadratic; ||S||<1)
    float* Xc = Xa;
    float* Xn = Xb;
    const dim3 gB(BLKD / WG_N, BLKD / WG_M, NBLK);   // (16, 8, 4)
    for (int it = 0; it < 5; ++it) {
        // T = (I + S) * Xc
        wmma_gemm_f32<<<gB, thr, 0, stream>>>(S, Xc, T,
                                              BLKD, BLKD, BLKD, BLKD, BLKD, BLKD,
                                              BB, BB, BB, /*aMode=*/1, 0, 0);
        // Xn = Xc * (2I - T)
        wmma_gemm_f32<<<gB, thr, 0, stream>>>(Xc, T, Xn,
                                              BLKD, BLKD, BLKD, BLKD, BLKD, BLKD,
                                              BB, BB, BB, 0, /*bMode=*/1, 0);
        float* tmp = Xc; Xc = Xn; Xn = tmp;
    }
    // Q = (I - S) * X   -> T
    wmma_gemm_f32<<<gB, thr, 0, stream>>>(S, Xc, T,
                                          BLKD, BLKD, BLKD, BLKD, BLKD, BLKD,
                                          BB, BB, BB, /*aMode=*/2, 0, 0);

    // filt_t[r*B+i][o] = sum_j Q_r[i][j] * W[o][r*B+j]   (B read transposed)
    const dim3 gF(OUT_F / WG_N, BLKD / WG_M, NBLK);  // (64, 8, 4)
    wmma_gemm_f32<<<gF, thr, 0, stream>>>(T, W, filt,
                                          BLKD, OUT_F, BLKD,
                                          BLKD, IN_F, OUT_F,
                                          BB, (long)BLKD, (long)BLKD * OUT_F,
                                          0, 0, /*bTrans=*/1);

    // out = x @ filt_t  (8192 x 4096 x 4096) — async-LDS double-buffered path
    const dim3 gO(OUT_F / WG_N, MX / WG_M, 1);       // (64, 64, 1)
    wmma_gemm_f32_async<<<gO, thr, 0, stream>>>(x, filt, out,
                                                MX, OUT_F, IN_F,
                                                IN_F, OUT_F, OUT_F);
}